// Sampler_29652454212392
// MI455X (gfx1250) — compile-verified
//
#include <hip/hip_runtime.h>
#include <cstdint>

#define BLOCK 1024         // 32 wave32 waves: max waves/WG, max MLP per WGP
#define NWAVES (BLOCK / 32)
#define TILE (BLOCK * 4)   // floats per LDS tile (each thread owns 16B)

// ---- monotone uint32 key for float ordering (ascending float -> ascending uint) ----
__device__ __forceinline__ unsigned fkey(float f) {
  unsigned u = __float_as_uint(f);
  return (u & 0x80000000u) ? ~u : (u | 0x80000000u);
}

// ---- wave32 reductions ----
__device__ __forceinline__ void wave_argmax(float& v, int& i) {
#pragma unroll
  for (int off = 16; off > 0; off >>= 1) {
    float ov = __shfl_xor(v, off, 32);
    int oi = __shfl_xor(i, off, 32);
    if (ov > v || (ov == v && oi < i)) { v = ov; i = oi; }
  }
}

// ---- CDNA5 async global -> LDS copy (16B per lane, one instruction per wave) ----
__device__ __forceinline__ void async_tile_b128(const float* g, float* l) {
#if defined(__HIP_DEVICE_COMPILE__) && defined(__gfx1250__)
  // Low 32 bits of the generic shared-memory address are the LDS byte offset
  // (ISA 10.2 aperture table).
  unsigned lo = (unsigned)(size_t)l;
  asm volatile("global_load_async_to_lds_b128 %0, %1, off"
               :: "v"(lo), "v"(g) : "memory");
#else
  *(float4*)l = *(const float4*)g;
#endif
}

__device__ __forceinline__ void wait_async_le(int n) {
#if defined(__HIP_DEVICE_COMPILE__) && defined(__gfx1250__)
  if (n == 0) asm volatile("s_wait_asynccnt 0" ::: "memory");
  else        asm volatile("s_wait_asynccnt 1" ::: "memory");
#else
  (void)n;
#endif
}

__global__ __launch_bounds__(BLOCK) void sampler_kernel(
    const float* __restrict__ logits, const float* __restrict__ temperature,
    const float* __restrict__ min_p, const int* __restrict__ top_k,
    const float* __restrict__ top_p, const float* __restrict__ q,
    int* __restrict__ out, int V) {
  const int row = blockIdx.x;
  const int tid = threadIdx.x;
  const int wave = tid >> 5, lane = tid & 31;
  const float* __restrict__ rowL = logits + (size_t)row * V;
  const float* __restrict__ rowQ = q + (size_t)row * V;

  __shared__ float s_tile[2][TILE];          // 32 KB async staging double buffer
  __shared__ float s_rv[NWAVES];
  __shared__ int s_ri[NWAVES];
  __shared__ unsigned s_cnt[256];            // top-k count histogram
  __shared__ unsigned long long s_mass[256]; // top-p mass histogram (2^32 fixed pt)
  __shared__ unsigned s_scanC[256];          // suffix scans
  __shared__ unsigned long long s_scanM[256];
  __shared__ float s_m;
  __shared__ int s_greedy;
  __shared__ unsigned s_prefK, s_prefP, s_kRem, s_kRemNext, s_foundP;
  __shared__ unsigned long long s_pRem, s_pRemNext;

  // Stream the q row (only used in the last pass) into L2 behind the logit phases.
  for (int i = tid * 32; i < V; i += BLOCK * 32)
    __builtin_prefetch(rowQ + i, 0, 1);      // global_prefetch_b8, no counters

  const float tRaw = temperature[row];
  const float tAdj = (tRaw < 1e-5f) ? 1.0f : tRaw;
  const float invT = 1.0f / tAdj;

  // ============ Phase A: max + argmax (greedy), async-LDS staged ==============
  float bv = -__builtin_inff();
  int bi = 0x7fffffff;
  const int nTiles = (V + TILE - 1) / TILE;
  {
    int base0 = tid * 4;
    if (base0 < V) async_tile_b128(rowL + base0, &s_tile[0][tid * 4]);
    for (int t = 0; t < nTiles; ++t) {
      int nb = t + 1;
      if (nb < nTiles) {
        int base = nb * TILE + tid * 4;
        if (base < V) async_tile_b128(rowL + base, &s_tile[nb & 1][tid * 4]);
        wait_async_le(1);   // own tile-t copy done; tile t+1 still in flight
      } else {
        wait_async_le(0);
      }
      __syncthreads();      // all waves' tile-t data resident in LDS
      int base = t * TILE + tid * 4;
      if (base < V) {
        float4 vv = *(const float4*)&s_tile[t & 1][tid * 4];
        if (vv.x > bv) { bv = vv.x; bi = base + 0; }
        if (vv.y > bv) { bv = vv.y; bi = base + 1; }
        if (vv.z > bv) { bv = vv.z; bi = base + 2; }
        if (vv.w > bv) { bv = vv.w; bi = base + 3; }
      }
      __syncthreads();      // done reading before this buffer is overwritten
    }
  }
  wave_argmax(bv, bi);
  if (lane == 0) { s_rv[wave] = bv; s_ri[wave] = bi; }
  __syncthreads();
  if (wave == 0) {
    float v = (lane < NWAVES) ? s_rv[lane] : -__builtin_inff();
    int i = (lane < NWAVES) ? s_ri[lane] : 0x7fffffff;
    wave_argmax(v, i);
    if (lane == 0) { s_m = v; s_greedy = i; }
  }
  if (tid == 0) { s_prefK = 0u; s_prefP = 0u; s_kRem = (unsigned)top_k[row]; }
  __syncthreads();

  const float mprime = s_m * invT;                 // max of scaled logits
  const float thr = mprime + __logf(min_p[row]);   // min-p logit threshold (-inf ok)

  // ==== Phase B: dual radix-select; round 0 mass total doubles as Z ===========
  for (int shift = 24; shift >= 0; shift -= 8) {
    if (tid < 256) { s_cnt[tid] = 0u; s_mass[tid] = 0ull; }
    __syncthreads();
    const unsigned maskHi = (shift == 24) ? 0u : (0xFFFFFFFFu << (shift + 8));
    const unsigned pk = s_prefK, pp = s_prefP;
    for (int i = tid * 4; i < V; i += TILE) {
      float4 vv = *(const float4*)(rowL + i);
      float l[4] = {vv.x, vv.y, vv.z, vv.w};
#pragma unroll
      for (int j = 0; j < 4; ++j) {
        float lp = l[j] * invT;
        bool kept = (lp >= thr);
        unsigned key = kept ? fkey(lp) : 0u;   // masked -> below every finite key
        unsigned bin = (key >> shift) & 255u;
        if ((key & maskHi) == pk) atomicAdd(&s_cnt[bin], 1u);       // ds_add_u32
        if (kept && ((key & maskHi) == pp)) {
          // 2^32 fixed point: exact, commutative -> deterministic replay
          unsigned long long fx =
              (unsigned long long)(__expf(lp - mprime) * 4294967296.0f);
          atomicAdd(&s_mass[bin], fx);                              // ds_add_u64
        }
      }
    }
    __syncthreads();
    // ---- parallel inclusive suffix scan over 256 bins (8 log-steps) ----
    if (tid < 256) { s_scanC[tid] = s_cnt[tid]; s_scanM[tid] = s_mass[tid]; }
    __syncthreads();
#pragma unroll
    for (int d = 1; d < 256; d <<= 1) {
      unsigned ca = 0; unsigned long long ma = 0ull;
      if (tid < 256 - d) { ca = s_scanC[tid + d]; ma = s_scanM[tid + d]; }
      __syncthreads();
      if (tid < 256) { s_scanC[tid] += ca; s_scanM[tid] += ma; }
      __syncthreads();
    }
    if (tid == 0) s_foundP = 0u;
    __syncthreads();
    // ---- crossing detection: one thread per bin ----
    if (tid < 256) {
      const unsigned krem = s_kRem;
      unsigned long long prem;
      if (shift == 24) {  // round 0: total mass == Z (trivial prefix)
        double t = (double)top_p[row] * (double)s_scanM[0];
        prem = (unsigned long long)t;
        if (prem == 0ull) prem = 1ull;   // p==0 -> keep only max-prob ties
        if (tid == 0) s_pRem = prem;     // publish target (also covers fallback)
      } else {
        prem = s_pRem;
      }
      unsigned sc = s_scanC[tid];
      unsigned scn = (tid < 255) ? s_scanC[tid + 1] : 0u;
      if (sc >= krem && scn < krem) {    // exactly one bin satisfies this
        s_prefK |= ((unsigned)tid) << shift;
        s_kRemNext = krem - scn;
      }
      unsigned long long sm = s_scanM[tid];
      unsigned long long smn = (tid < 255) ? s_scanM[tid + 1] : 0ull;
      if (sm >= prem && smn < prem) {
        s_prefP |= ((unsigned)tid) << shift;
        s_pRemNext = prem - smn;
        s_foundP = 1u;
      }
    }
    __syncthreads();
    if (tid == 0) {
      s_kRem = s_kRemNext;
      if (s_foundP) {
        s_pRem = s_pRemNext;
      } else {  // target exceeds total mass (p ~= 1): keep lowest non-empty bin
        int sel = 0;
        for (int b = 0; b < 256; ++b)
          if (s_mass[b] != 0ull) { sel = b; break; }
        s_prefP |= ((unsigned)sel) << shift;
        // s_pRem unchanged: next level's total < target again -> fallback again,
        // terminating at the lowest non-empty leaf (keeps whole kept set).
      }
    }
    __syncthreads();
  }
  const unsigned tk = s_prefK, tp = s_prefP;   // exact cutoff keys (tie-inclusive)

  // ===== Phase C: argmax over kept set of l' - log(q)  (== argmax p/q) ========
  float sv = -__builtin_inff();
  int si = 0x7fffffff;
  for (int i = tid * 4; i < V; i += TILE) {
    float4 lv = *(const float4*)(rowL + i);
    float4 qv = *(const float4*)(rowQ + i);
    float l[4] = {lv.x, lv.y, lv.z, lv.w};
    float qq[4] = {qv.x, qv.y, qv.z, qv.w};
#pragma unroll
    for (int j = 0; j < 4; ++j) {
      float lp = l[j] * invT;
      if (lp >= thr) {
        unsigned key = fkey(lp);
        if (key >= tk && key >= tp) {
          float sc = lp - __logf(qq[j]);
          if (sc > sv) { sv = sc; si = i + j; }
        }
      }
    }
  }
  wave_argmax(sv, si);
  if (lane == 0) { s_rv[wave] = sv; s_ri[wave] = si; }
  __syncthreads();
  if (wave == 0) {
    float v = (lane < NWAVES) ? s_rv[lane] : -__builtin_inff();
    int i2 = (lane < NWAVES) ? s_ri[lane] : 0x7fffffff;
    wave_argmax(v, i2);
    if (lane == 0) out[row] = (tRaw < 1e-5f) ? s_greedy : i2;
  }
}

extern "C" void kernel_launch(void* const* d_in, const int* in_sizes, int n_in,
                              void* d_out, int out_size, void* d_ws, size_t ws_size,
                              hipStream_t stream) {
  const float* logits      = (const float*)d_in[0];
  const float* temperature = (const float*)d_in[1];
  const float* min_p       = (const float*)d_in[2];
  const int*   top_k       = (const int*)d_in[3];
  const float* top_p       = (const float*)d_in[4];
  const float* q           = (const float*)d_in[5];
  const int B = in_sizes[1];              // temperature is [B]
  const int V = in_sizes[0] / B;          // logits is [B, V]
  sampler_kernel<<<B, BLOCK, 0, stream>>>(logits, temperature, min_p, top_k,
                                          top_p, q, (int*)d_out, V);
}